// SA_61272003445141
// MI455X (gfx1250) — compile-verified
//
#include <hip/hip_runtime.h>

typedef float v2f __attribute__((ext_vector_type(2)));
typedef float v8f __attribute__((ext_vector_type(8)));

static constexpr int kD = 256;      // feature dim
static constexpr int kL = 4096;     // sequence length
static constexpr int kB = 4;        // batch
static constexpr int kRows = kB * kL;   // 16384 flattened (b,l) rows

// Workspace layout (float offsets)
static constexpr size_t WS_WQ = 0;                  // wq_eff[256]
static constexpr size_t WS_WK = 256;                // wk_eff[256]
static constexpr size_t WS_QS = 512;                // q_s[16384]
static constexpr size_t WS_KS = WS_QS + kRows;      // k_s[16384]
static constexpr size_t WS_V  = WS_KS + kRows;      // V[16384*256] (16B-aligned: 133120 B)

// ---------------------------------------------------------------------------
// Kernel A: wq_eff[d] = sum_e Wq[e][d]*w[e] ; same for Wk. One 256-thread block.
// ---------------------------------------------------------------------------
__global__ void SA_proj_vec(const float* __restrict__ Wq, const float* __restrict__ Wk,
                            const float* __restrict__ w, float* __restrict__ ws) {
    int d = threadIdx.x;
    float aq = 0.f, ak = 0.f;
#pragma unroll 4
    for (int e = 0; e < kD; ++e) {
        float we = w[e];
        aq += Wq[e * kD + d] * we;   // coalesced: lanes walk d
        ak += Wk[e * kD + d] * we;
    }
    ws[WS_WQ + d] = aq;
    ws[WS_WK + d] = ak;
}

// ---------------------------------------------------------------------------
// Kernel B: q_s[row] = x[row,:]·wq_eff ; k_s[row] = x[row,:]·wk_eff.
// One wave (32 lanes) per row, shuffle reduction. 8 rows per 256-thread block.
// ---------------------------------------------------------------------------
__global__ void SA_row_scores(const float* __restrict__ x, float* __restrict__ ws) {
    const float* wq = ws + WS_WQ;
    const float* wk = ws + WS_WK;
    int wave = threadIdx.x >> 5;
    int lane = threadIdx.x & 31;
    int row  = blockIdx.x * 8 + wave;
    const float* xr = x + (size_t)row * kD;
    float sq = 0.f, sk = 0.f;
#pragma unroll
    for (int t = 0; t < kD / 32; ++t) {
        float xv = xr[lane + 32 * t];          // coalesced
        sq += xv * wq[lane + 32 * t];
        sk += xv * wk[lane + 32 * t];
    }
#pragma unroll
    for (int off = 16; off; off >>= 1) {       // wave32 reduction
        sq += __shfl_xor(sq, off, 32);
        sk += __shfl_xor(sk, off, 32);
    }
    if (lane == 0) {
        ws[WS_QS + row] = sq;
        ws[WS_KS + row] = sk;
    }
}

// ---------------------------------------------------------------------------
// Kernel C: V = x @ Wv^T via V_WMMA_F32_16X16X4_F32.
// Block = 256 thr = 8 waves handles a 16-row tile; wave w owns column tiles
// {16w, 16w+128}. x tile staged in LDS with 260-float pitch (bank-conflict-free
// float2 reads: 260 ≡ 4 mod 64 → distinct banks for m=0..15).
// ---------------------------------------------------------------------------
__global__ void SA_vproj(const float* __restrict__ x, const float* __restrict__ Wv,
                         float* __restrict__ ws) {
    __shared__ __align__(16) float xs[16 * 260];
    float* V = ws + WS_V;
    int row0 = blockIdx.x * 16;
    int tid  = threadIdx.x;

    {   // stage 16x256 x tile: each thread moves 16 floats as 4x float4
        int m = tid >> 4, c = (tid & 15) * 16;
        const float4* src = (const float4*)(x + (size_t)(row0 + m) * kD + c);
        float4* dst = (float4*)(&xs[m * 260 + c]);   // 260 % 4 == 0 → 16B aligned
#pragma unroll
        for (int q = 0; q < 4; ++q) dst[q] = src[q];
    }
    __syncthreads();

    int wave = tid >> 5, lane = tid & 31;
    int ln = lane & 15, half = lane >> 4;
    int col0a = wave * 16, col0b = wave * 16 + 128;
    v8f acc0 = {}, acc1 = {};

#pragma unroll 4
    for (int kb = 0; kb < kD; kb += 4) {
        v2f a = *(const v2f*)(&xs[ln * 260 + kb + 2 * half]);
        v2f b0 = *(const v2f*)(Wv + (size_t)(col0a + ln) * kD + kb + 2 * half);
        v2f b1 = *(const v2f*)(Wv + (size_t)(col0b + ln) * kD + kb + 2 * half);
        acc0 = __builtin_amdgcn_wmma_f32_16x16x4_f32(false, a, false, b0,
                                                     (short)0, acc0, false, false);
        acc1 = __builtin_amdgcn_wmma_f32_16x16x4_f32(false, a, false, b1,
                                                     (short)0, acc1, false, false);
    }
    // C/D layout: VGPR r → M = r + 8*half, N = lane&15
#pragma unroll
    for (int r = 0; r < 8; ++r) {
        int rr = row0 + r + 8 * half;
        V[(size_t)rr * kD + col0a + ln] = acc0[r];
        V[(size_t)rr * kD + col0b + ln] = acc1[r];
    }
}

// ---------------------------------------------------------------------------
// Kernel D: attention. One block (8 waves) per (batch, 16-query tile).
// V key tiles (16x256 f32 = 16KB) are double-buffered in LDS via
// GLOBAL_LOAD_ASYNC_TO_LDS_B128 (ASYNCcnt), prefetched one tile ahead so the
// L2 fetch overlaps exp + WMMA of the current tile. P tile built in LDS
// (pitch 18 → conflict-free A-frag float2 reads). V-tile pitch 272 floats:
// the two key rows of a B-frag are 544 dwords ≡ 32 banks apart and half-waves
// span disjoint bank ranges → conflict-free ds_load_b32.
// No running softmax max needed: scores ≥ 0 and bounded (~10) → exp sums are
// safely inside fp32 range (softmax is shift-invariant; identical math).
// ---------------------------------------------------------------------------
__global__ void SA_attn(const float* __restrict__ ws, const float* __restrict__ bptr,
                        float* __restrict__ out) {
    constexpr int PITCH = 272;                          // floats
    __shared__ __align__(16) float vt[2][16 * PITCH];   // double-buffered V tile
    __shared__ float p[16 * 18];
    __shared__ float qs_s[16];
    __shared__ float inv_s[16];
    const float* qs = ws + WS_QS;
    const float* ks = ws + WS_KS;
    const float* V  = ws + WS_V;

    int b  = blockIdx.x >> 8;            // 256 query tiles per batch
    int i0 = (blockIdx.x & 255) * 16;
    int tid = threadIdx.x;
    int wave = tid >> 5, lane = tid & 31;
    int mn = lane & 15, half = lane >> 4;
    int pm = tid & 15, pk = tid >> 4;    // (row, key) this thread fills in P
    int skey = tid >> 4, schunk = tid & 15;  // staging: key row, 16-col chunk
    float bias = bptr[0];

    const float* Vb  = V + (size_t)b * kL * kD;
    const float* ksb = ks + (size_t)b * kL;

    // async-stage one 16-key V tile into LDS buffer `buf` (4x b128 per thread)
    auto stage = [&](int j0n, int buf) {
        const float* g = Vb + (size_t)(j0n + skey) * kD + schunk * 16;
        uint32_t l = (uint32_t)(size_t)(&vt[buf][skey * PITCH + schunk * 16]);
        asm volatile("global_load_async_to_lds_b128 %0, %1, off"
                     :: "v"(l),      "v"(g)      : "memory");
        asm volatile("global_load_async_to_lds_b128 %0, %1, off"
                     :: "v"(l + 16), "v"(g + 4)  : "memory");
        asm volatile("global_load_async_to_lds_b128 %0, %1, off"
                     :: "v"(l + 32), "v"(g + 8)  : "memory");
        asm volatile("global_load_async_to_lds_b128 %0, %1, off"
                     :: "v"(l + 48), "v"(g + 12) : "memory");
    };

    if (tid < 16) qs_s[tid] = qs[(size_t)b * kL + i0 + tid];
    stage(0, 0);                          // prologue: tile 0 -> buffer 0
    __syncthreads();                      // qs_s visible

    int col0a = wave * 16, col0b = wave * 16 + 128;
    v8f acc0 = {}, acc1 = {};
    float den = 0.f;                      // live in wave0 lanes 0..15 (row = tid)

    for (int j0 = 0; j0 < kL; j0 += 16) {
        int buf = (j0 >> 4) & 1;
        float pv = __expf(fmaxf(qs_s[pm] - ksb[j0 + pk] + bias, 0.f));

        asm volatile("s_wait_asynccnt 0x0" ::: "memory");  // own tile-copies done
        __syncthreads();          // all waves: prev P consumed + V tile complete
        p[pm * 18 + pk] = pv;
        __syncthreads();          // P tile ready

        // prefetch next V tile into the other buffer. Safe: that buffer's
        // readers (iteration j0-16) drained their DS reads before the first
        // barrier of THIS iteration (syncthreads waits DScnt).
        if (j0 + 16 < kL) stage(j0 + 16, buf ^ 1);

        if (tid < 16) {                   // denominator row-sums (wave 0 only)
            float s = 0.f;
#pragma unroll
            for (int k = 0; k < 16; ++k) s += p[tid * 18 + k];
            den += s;
        }

        const float* vb = &vt[buf][0];
        v2f a[4];
#pragma unroll
        for (int s = 0; s < 4; ++s)       // A-frags: K chunks of 4 keys
            a[s] = *(const v2f*)(&p[mn * 18 + 4 * s + 2 * half]);
#pragma unroll
        for (int s = 0; s < 4; ++s) {
            int k0 = 4 * s + 2 * half;
            v2f b0, b1;
            b0.x = vb[k0 * PITCH + col0a + mn];
            b0.y = vb[(k0 + 1) * PITCH + col0a + mn];
            b1.x = vb[k0 * PITCH + col0b + mn];
            b1.y = vb[(k0 + 1) * PITCH + col0b + mn];
            acc0 = __builtin_amdgcn_wmma_f32_16x16x4_f32(false, a[s], false, b0,
                                                         (short)0, acc0, false, false);
            acc1 = __builtin_amdgcn_wmma_f32_16x16x4_f32(false, a[s], false, b1,
                                                         (short)0, acc1, false, false);
        }
    }

    if (tid < 16) inv_s[tid] = 1.0f / den;
    __syncthreads();

    float* outb = out + (size_t)b * kL * kD;
#pragma unroll
    for (int r = 0; r < 8; ++r) {
        int rr = r + 8 * half;            // C layout: M = r + 8*half
        float inv = inv_s[rr];
        outb[(size_t)(i0 + rr) * kD + col0a + mn] = acc0[r] * inv;
        outb[(size_t)(i0 + rr) * kD + col0b + mn] = acc1[r] * inv;
    }
}

// ---------------------------------------------------------------------------
extern "C" void kernel_launch(void* const* d_in, const int* in_sizes, int n_in,
                              void* d_out, int out_size, void* d_ws, size_t ws_size,
                              hipStream_t stream) {
    const float* x   = (const float*)d_in[0];
    const float* Wq  = (const float*)d_in[1];
    const float* Wk  = (const float*)d_in[2];
    const float* Wv  = (const float*)d_in[3];
    const float* wml = (const float*)d_in[4];
    const float* bml = (const float*)d_in[5];
    float* out = (float*)d_out;
    float* ws  = (float*)d_ws;
    (void)in_sizes; (void)n_in; (void)out_size; (void)ws_size;

    SA_proj_vec  <<<1,          256, 0, stream>>>(Wq, Wk, wml, ws);
    SA_row_scores<<<kRows / 8,  256, 0, stream>>>(x, ws);
    SA_vproj     <<<kRows / 16, 256, 0, stream>>>(x, Wv, ws);
    SA_attn      <<<kB * (kL / 16), 256, 0, stream>>>(ws, bml, out);
}